// Prior_SPDE_85650237817232
// MI455X (gfx1250) — compile-verified
//
#include <hip/hip_runtime.h>

typedef float v2f __attribute__((ext_vector_type(2)));
typedef float v4f __attribute__((ext_vector_type(4)));
typedef float v8f __attribute__((ext_vector_type(8)));

#define NN  1024
#define NT  9
#define NCH 25

#define BW    96        // dense band width (delta range [-47,47] + pad)
#define BOFS  47        // delta offset into band
#define NROW  80        // stencil rows staged per workgroup
#define WOFF  (NROW*BW) // offset of w-scaled band copy

// U[i,k] = -w_i * invM_t[i,k], 5-point stencil with Dirichlet masks baked in.
__device__ __forceinline__ float stencil_u(const float* __restrict__ m,
                                           const float* __restrict__ tau,
                                           float kap2, int b, int t, int i, int k) {
  int d = k - i;
  bool hit = (d == 0) | (d == 1) | (d == -1) | (d == 32) | (d == -32);
  if (!hit) return 0.f;
  int gx = i & 31, gy = i >> 5;
  float ta = tau[((size_t)b * NN + i) * NT + t];
  float w = 1.f / (ta * ta);                       // DT = 1
  const float* m1 = m + (size_t)(b * 2 + 0) * NN * NT;
  const float* m2 = m + (size_t)(b * 2 + 1) * NN * NT;
  float a;
  if (d == 0)        a = 1.f + kap2 + 4.f;         // invM diag = 1 + kappa^2 + 4
  else if (d == 1)   a = (gx < 31) ? (-1.f + 0.5f * m1[i * NT + t]) : 0.f;
  else if (d == -1)  a = (gx > 0)  ? (-1.f - 0.5f * m1[i * NT + t]) : 0.f;
  else if (d == 32)  a = (gy < 31) ? (-1.f + 0.5f * m2[i * NT + t]) : 0.f;
  else               a = (gy > 0)  ? (-1.f - 0.5f * m2[i * NT + t]) : 0.f;
  return -w * a;
}

// Kernel 1: stream the full 210MB output with NT b128 stores.
// c in [0,9): zeros (D band overwritten by WMMA kernel afterwards)
// c in [9,17): L[tau=c-9] ; c in [17,25): U[tau=c-17]
__global__ void __launch_bounds__(256) fill_ul_kernel(
    const float* __restrict__ kappa, const float* __restrict__ m,
    const float* __restrict__ tau, float* __restrict__ out) {
  int gid = blockIdx.x;                 // flat (b, c, i) row id
  int i = gid & (NN - 1);
  int c = (gid >> 10) % NCH;
  int b = gid / (NN * NCH);
  int k0 = threadIdx.x * 4;
  float kap = kappa[0];
  float kap2 = kap * kap;
  v4f v = {0.f, 0.f, 0.f, 0.f};
  if (c >= 9) {
    #pragma unroll
    for (int j = 0; j < 4; ++j) {
      int k = k0 + j;
      float r;
      if (c < 17) r = stencil_u(m, tau, kap2, b, c - 8, k, i);   // L = U^T (t = c-9+1)
      else        r = stencil_u(m, tau, kap2, b, c - 16, i, k);  // U       (t = c-17+1)
      v[j] = r;
    }
  }
  __builtin_nontemporal_store(v, (v4f*)(out + (size_t)gid * NN + k0));
}

// Kernel 2: D channels (c=0..8) band tiles via V_WMMA_F32_16X16X4_F32.
// G(I,K) = sum_J M(J,I)^T diag(w_J) M(J,K); nonzero for |I-K|<=4, J in overlap of
// [I-2,I+2] and [K-2,K+2]. One wave32 per (b,c,I).
// LDS holds a dense zero-padded band per stencil row so every WMMA operand
// element is ONE unconditional ds_load_b32 (no selects, no EXEC divergence):
//   bands[r*BW + (j-n) + BOFS]        = M[n, j]
//   bands[WOFF + r*BW + (j-n) + BOFS] = w[n] * M[n, j]
__global__ void __launch_bounds__(32) band_wmma_kernel(
    const float* __restrict__ kappa, const float* __restrict__ m,
    const float* __restrict__ tau, float* __restrict__ out) {
  const int I    = blockIdx.x;          // block row, 0..63
  const int c    = blockIdx.y;          // D channel, 0..8
  const int b    = blockIdx.z;
  const int lane = threadIdx.x;         // wave32
  const int half = lane >> 4;
  const int lp   = lane & 15;
  const int t    = (c == 0) ? 0 : c;    // B0 uses A_0 ; invM[c-1] uses A_c

  __shared__ float bands[2 * NROW * BW];   // [0,WOFF): M ; [WOFF,..): w*M
  __shared__ float sw[NROW];               // w per row (diag add)

  // Zero-fill the bands with b128 LDS stores.
  {
    v4f z = {0.f, 0.f, 0.f, 0.f};
    v4f* b4 = (v4f*)bands;
    for (int idx = lane; idx < (2 * NROW * BW) / 4; idx += 32) b4[idx] = z;
  }
  __syncthreads();

  float kap = kappa[0];
  float kap2 = kap * kap;
  const float* m1 = m + (size_t)(b * 2 + 0) * NN * NT;
  const float* m2 = m + (size_t)(b * 2 + 1) * NN * NT;

  for (int r = lane; r < NROW; r += 32) {
    int n = 16 * I - 32 + r;
    float w = 0.f;
    if (n >= 0 && n < NN) {
      int gx = n & 31, gy = n >> 5;
      float a1 = m1[n * NT + t], a2 = m2[n * NT + t];
      float cD = kap2 + 4.f + ((c >= 1) ? 1.f : 0.f);   // +1 for invM = I + A
      float cE = (gx < 31) ? (-1.f + 0.5f * a1) : 0.f;
      float cW = (gx > 0)  ? (-1.f - 0.5f * a1) : 0.f;
      float cN = (gy < 31) ? (-1.f + 0.5f * a2) : 0.f;
      float cS = (gy > 0)  ? (-1.f - 0.5f * a2) : 0.f;
      if (c == 0) w = 1.f;                              // Q0: DX*DY = 1
      else { float ta = tau[((size_t)b * NN + n) * NT + t]; w = 1.f / (ta * ta); }
      float* row  = &bands[r * BW + BOFS];
      float* rowW = &bands[WOFF + r * BW + BOFS];
      row[-32] = cS;      rowW[-32] = w * cS;
      row[-1]  = cW;      rowW[-1]  = w * cW;
      row[0]   = cD;      rowW[0]   = w * cD;
      row[1]   = cE;      rowW[1]   = w * cE;
      row[32]  = cN;      rowW[32]  = w * cN;
    }
    sw[r] = w;
  }
  __syncthreads();

  for (int K = I - 4; K <= I + 4; ++K) {
    if (K < 0 || K > 63) continue;
    v8f acc = {};
    int Jlo = ((I > K) ? I : K) - 2; if (Jlo < 0)  Jlo = 0;
    int Jhi = ((I < K) ? I : K) + 2; if (Jhi > 63) Jhi = 63;
    for (int J = Jlo; J <= Jhi; ++J) {
      #pragma unroll
      for (int q = 0; q < 4; ++q) {
        // A operand layout (16x4 f32): lanes<16 hold K={0,1}, lanes>=16 K={2,3}
        int noff = 4 * q + 2 * half;                 // contraction row within J tile
        int r    = 16 * (J - I) + 32 + noff;         // LDS row index (0..78)
        int dA   = 16 * (I - J) + lp - noff;         // A elem = M[n, 16I+lp] (transposed use)
        int dB   = 16 * (K - J) + lp - noff;         // B elem = w[n]*M[n, 16K+lp]
        int ra   = r * BW + dA + BOFS;               // in [1, 94]
        int rb   = WOFF + r * BW + dB + BOFS;
        v2f a, bb;
        a.x  = bands[ra];
        a.y  = bands[ra + BW - 1];                   // row r+1, delta dA-1
        bb.x = bands[rb];
        bb.y = bands[rb + BW - 1];                   // row r+1, delta dB-1
        acc = __builtin_amdgcn_wmma_f32_16x16x4_f32(
            /*neg_a=*/false, a, /*neg_b=*/false, bb,
            /*c_mod=*/(short)0, acc, /*reuse_a=*/false, /*reuse_b=*/false);
      }
    }
    // Diagonal terms: D0 += 1.05*I (Q0's 0.05I + I); D_mid += diag(ws); D_last none.
    if (K == I && c <= 7) {
      int v = lp - 8 * half;        // C layout: elem (i=v+8*half, k=lp); diag i==k
      if (v >= 0 && v < 8) {
        float dterm = (c == 0) ? 1.05f : sw[lp + 32];
        acc[v] += dterm;
      }
    }
    size_t base = (((size_t)(b * NCH + c)) * NN + 16 * I) * (size_t)NN + 16 * K;
    #pragma unroll
    for (int v = 0; v < 8; ++v) {
      int ii = v + 8 * half;
      out[base + (size_t)ii * NN + lp] = acc[v];
    }
  }
}

extern "C" void kernel_launch(void* const* d_in, const int* in_sizes, int n_in,
                              void* d_out, int out_size, void* d_ws, size_t ws_size,
                              hipStream_t stream) {
  const float* kappa = (const float*)d_in[0];
  const float* m     = (const float*)d_in[1];
  // d_in[2] = H (unused by reference)
  const float* tau   = (const float*)d_in[3];
  float* out = (float*)d_out;

  // 1) Stream the full 2*25*1024*1024 output: zeros + banded L/U values.
  fill_ul_kernel<<<dim3(2 * NCH * NN), dim3(256), 0, stream>>>(kappa, m, tau, out);

  // 2) Overwrite the 13-band D tiles via fp32 WMMA Gram products.
  band_wmma_kernel<<<dim3(64, 9, 2), dim3(32), 0, stream>>>(kappa, m, tau, out);
}